// Triaffine_61125974557035
// MI455X (gfx1250) — compile-verified
//
#include <hip/hip_runtime.h>

typedef float v2f __attribute__((ext_vector_type(2)));
typedef float v8f __attribute__((ext_vector_type(8)));

#define B_ 8
#define L_ 256
#define Z_ 16
#define H_ 256
#define TPAD 260  // 16-row T tile, padded row stride: (4*row+col)%64 -> conflict-free

// ---------------------------------------------------------------------------
// Kernel 1: WC[bz, i, j] = sum_o cls[bz, o] * W[i, o, j]
// GEMM: M = B*Z = 128 (8 m-tiles, all done by each wave, sharing B-fragment),
//       N = H*H = 65536 (one 16-wide (i,j) tile per wave), K = H = 256.
// ---------------------------------------------------------------------------
__global__ __launch_bounds__(128) void triaffine_wc(
    const float* __restrict__ cls,   // [B*Z, H]
    const float* __restrict__ W,     // [H, H, H]  (i, o, j)
    float* __restrict__ WC)          // [B*Z, H*H] (bz, i*H + j)
{
    const int lane = threadIdx.x & 31;
    const int wave = threadIdx.x >> 5;
    const int tile = blockIdx.x * 4 + wave;   // 0..4095, n0 = tile*16
    const int i    = tile >> 4;               // all 16 columns share i
    const int j0   = (tile & 15) << 4;
    const int l15  = lane & 15;
    const int hi   = lane >> 4;               // 0: K=k0,k0+1   1: K=k0+2,k0+3

    v8f acc[8] = {};

    const float* Wcol = W + (i * H_) * H_ + j0 + l15;  // + k*H_ selects o-row k

    for (int k0 = 0; k0 < H_; k0 += 4) {
        const int kb = k0 + 2 * hi;
        v2f bf;
        bf.x = Wcol[kb * H_];
        bf.y = Wcol[(kb + 1) * H_];
        #pragma unroll
        for (int m = 0; m < 8; ++m) {
            v2f af = *(const v2f*)(cls + (m * 16 + l15) * H_ + kb);
            acc[m] = __builtin_amdgcn_wmma_f32_16x16x4_f32(
                false, af, false, bf, (short)0, acc[m], false, false);
        }
    }

    #pragma unroll
    for (int m = 0; m < 8; ++m) {
        const int rowbase = m * 16 + hi * 8;
        #pragma unroll
        for (int r = 0; r < 8; ++r)
            WC[(rowbase + r) * (H_ * H_) + (tile << 4) + l15] = acc[m][r];
    }
}

// ---------------------------------------------------------------------------
// Kernel 2 (fused stages 2+3): one WG per (b, z, 16-wide x tile).
//   Stage B: T[16, 256] = start[b, x0:x0+16, :] @ WC[bz]   (contract i) -> LDS
//   Stage C: out[b, x, y, z] = T @ end[b]^T                (contract j)
// 8 waves; each wave computes two 16-wide N tiles per stage, sharing A frags.
// ---------------------------------------------------------------------------
__global__ __launch_bounds__(256) void triaffine_span(
    const float* __restrict__ start, // [B, L, H]
    const float* __restrict__ endL,  // [B, L, H]
    const float* __restrict__ WC,    // [B*Z, H, H]
    float* __restrict__ out)         // [B, L, L, Z]
{
    __shared__ float T[16 * TPAD];

    const int lane = threadIdx.x & 31;
    const int wave = threadIdx.x >> 5;
    const int l15  = lane & 15;
    const int hi   = lane >> 4;

    const int gid = blockIdx.x;        // 0..2047
    const int xt  = gid & 15;
    const int z   = (gid >> 4) & 15;
    const int b   = gid >> 8;
    const int x0  = xt << 4;
    const int bz  = b * Z_ + z;

    // ---- Stage B: fill T tile in LDS -------------------------------------
    {
        const int j0 = wave * 32;                         // tiles j0, j0+16
        const float* aRow  = start + (b * L_ + x0 + l15) * H_;
        const float* WCc0  = WC + bz * (H_ * H_) + j0 + l15;
        const float* WCc1  = WCc0 + 16;
        v8f acc0 = {}, acc1 = {};
        for (int k0 = 0; k0 < H_; k0 += 4) {
            const int kb = k0 + 2 * hi;
            v2f af = *(const v2f*)(aRow + kb);
            v2f b0, b1;
            b0.x = WCc0[kb * H_];  b0.y = WCc0[(kb + 1) * H_];
            b1.x = WCc1[kb * H_];  b1.y = WCc1[(kb + 1) * H_];
            acc0 = __builtin_amdgcn_wmma_f32_16x16x4_f32(
                false, af, false, b0, (short)0, acc0, false, false);
            acc1 = __builtin_amdgcn_wmma_f32_16x16x4_f32(
                false, af, false, b1, (short)0, acc1, false, false);
        }
        #pragma unroll
        for (int r = 0; r < 8; ++r) {
            const int row = r + hi * 8;
            T[row * TPAD + j0 + l15]      = acc0[r];
            T[row * TPAD + j0 + 16 + l15] = acc1[r];
        }
    }
    __syncthreads();

    // ---- Stage C: out tile ------------------------------------------------
    {
        const int y0 = wave * 32;                         // tiles y0, y0+16
        const float* eR0  = endL + (b * L_ + y0 + l15) * H_;
        const float* eR1  = eR0 + 16 * H_;
        const float* Trow = T + l15 * TPAD;               // A from LDS (conflict-free)
        v8f acc0 = {}, acc1 = {};
        for (int k0 = 0; k0 < H_; k0 += 4) {
            const int kb = k0 + 2 * hi;
            v2f af = *(const v2f*)(Trow + kb);
            v2f b0 = *(const v2f*)(eR0 + kb);
            v2f b1 = *(const v2f*)(eR1 + kb);
            acc0 = __builtin_amdgcn_wmma_f32_16x16x4_f32(
                false, af, false, b0, (short)0, acc0, false, false);
            acc1 = __builtin_amdgcn_wmma_f32_16x16x4_f32(
                false, af, false, b1, (short)0, acc1, false, false);
        }
        float* ob = out + ((b * L_ + x0) * L_) * Z_ + z;
        #pragma unroll
        for (int r = 0; r < 8; ++r) {
            const int x = r + hi * 8;
            ob[(x * L_ + y0 + l15) * Z_]      = acc0[r];
            ob[(x * L_ + y0 + 16 + l15) * Z_] = acc1[r];
        }
    }
}

extern "C" void kernel_launch(void* const* d_in, const int* in_sizes, int n_in,
                              void* d_out, int out_size, void* d_ws, size_t ws_size,
                              hipStream_t stream) {
    const float* start = (const float*)d_in[0];   // [B, L, H]
    const float* endL  = (const float*)d_in[1];   // [B, L, H]
    const float* cls   = (const float*)d_in[2];   // [B, Z, H]
    const float* W     = (const float*)d_in[3];   // [H, H, H]
    float* out = (float*)d_out;                   // [B, L, L, Z]
    float* WC  = (float*)d_ws;                    // needs B*Z*H*H*4 = 32 MiB

    // Stage 1: WC = cls x W (contract o). 4096 N-tiles, 4 waves per block.
    triaffine_wc<<<1024, 128, 0, stream>>>(cls, W, WC);
    // Stages 2+3 fused through LDS. One WG per (b, z, x-tile): 8*16*16 = 2048.
    triaffine_span<<<2048, 256, 0, stream>>>(start, endL, WC, out);
}